// MatchingModule_27917287424588
// MI455X (gfx1250) — compile-verified
//
#include <hip/hip_runtime.h>

typedef float v2f __attribute__((ext_vector_type(2)));
typedef float v8f __attribute__((ext_vector_type(8)));

#define HID 768
#define SEQ 512
#define BATCH 64
#define WAVES_PER_BLOCK 8
#define BLOCK_THREADS (WAVES_PER_BLOCK * 32)
#define TILES_TOTAL (BATCH * SEQ / 16)   // 2048
#define GRID_BLOCKS (TILES_TOTAL / WAVES_PER_BLOCK)  // 256

// ---------------------------------------------------------------------------
// Kernel 1: cls_part[b][dir][c] = cls[b] @ W[:H][:,c] + bias[c]
// 256 threads = 64 batches x 2 dirs x 2 classes. Tiny (~0.4 MB reads).
// ---------------------------------------------------------------------------
__global__ void __launch_bounds__(256)
cls_base_kernel(const float* __restrict__ a2o_cls,
                const float* __restrict__ o2a_cls,
                const float* __restrict__ W_a2o, const float* __restrict__ b_a2o,
                const float* __restrict__ W_o2a, const float* __restrict__ b_o2a,
                float* __restrict__ base /* [B][4] = a0 a1 o0 o1 */) {
    int t   = threadIdx.x;      // 0..255
    int b   = t >> 2;
    int idx = t & 3;            // dir*2 + c
    int dir = idx >> 1;
    int c   = idx & 1;
    const float* cls  = dir ? o2a_cls : a2o_cls;
    const float* W    = dir ? W_o2a   : W_a2o;
    const float* bias = dir ? b_o2a   : b_a2o;
    const float* row = cls + (size_t)b * HID;
    float acc = bias[c];
    for (int h = 0; h < HID; ++h)
        acc = __builtin_fmaf(row[h], W[h * 2 + c], acc);
    base[b * 4 + idx] = acc;
}

// ---------------------------------------------------------------------------
// Kernel 2: main streaming kernel. One wave32 per 16-token tile.
// Uses V_WMMA_F32_16X16X4_F32 to compute both directions' opi @ W[H:] dots.
// ---------------------------------------------------------------------------
__global__ void __launch_bounds__(BLOCK_THREADS)
matching_wmma_kernel(const float* __restrict__ a2o_hid,   // [B,S,H]
                     const float* __restrict__ o2a_hid,   // [B,S,H]
                     const int*   __restrict__ mask_a,    // [B,S]
                     const int*   __restrict__ mask_o,    // [B,S]
                     const int*   __restrict__ ign,       // [B,S]
                     const float* __restrict__ W_a2o,     // [2H,2]
                     const float* __restrict__ W_o2a,     // [2H,2]
                     const float* __restrict__ base,      // [B][4]
                     float* __restrict__ out_logits,      // [B,S,2]
                     float* __restrict__ out_labels) {    // [B,S]
    // LDS: W[H:] transposed per direction -> wB[dir][c][k] = W[(H+k)*2 + c]
    __shared__ float wB[2][2][HID];                       // 12 KB
    __shared__ float ep[WAVES_PER_BLOCK][2][16][2];       // 2 KB epilogue bounce

    for (int i = threadIdx.x; i < 2 * HID; i += BLOCK_THREADS) {
        int k = i >> 1;
        int c = i & 1;
        wB[0][c][k] = W_a2o[(HID + k) * 2 + c];
        wB[1][c][k] = W_o2a[(HID + k) * 2 + c];
    }
    __syncthreads();

    const int lane  = threadIdx.x & 31;
    const int wave  = threadIdx.x >> 5;
    const int tile  = blockIdx.x * WAVES_PER_BLOCK + wave;
    const int b     = tile / (SEQ / 16);
    const int tok0  = (tile % (SEQ / 16)) * 16;

    const int row   = lane & 15;     // matrix row (token within tile) / N col
    const int khalf = lane >> 4;     // 0: K pair {0,1}; 1: K pair {2,3}
    const int nsel  = row & 1;       // lanes with N>=2 duplicate col N&1 (junk
                                     // D columns are never read)

    const float* pa = a2o_hid + ((size_t)(b * SEQ + tok0 + row)) * HID + khalf * 2;
    const float* po = o2a_hid + ((size_t)(b * SEQ + tok0 + row)) * HID + khalf * 2;
    const float* wa = &wB[0][nsel][khalf * 2];
    const float* wo = &wB[1][nsel][khalf * 2];

    v8f accA = {};
    v8f accO = {};

#pragma unroll 4
    for (int k = 0; k < HID; k += 4) {
        v2f aA = *(const v2f*)(pa + k);   // global_load_b64, cacheline-streamed
        v2f aO = *(const v2f*)(po + k);
        v2f bA = *(const v2f*)(wa + k);   // ds_load_b64 from staged W
        v2f bO = *(const v2f*)(wo + k);
        // D = A(16x4 f32) x B(4x16 f32) + C, full-precision accumulate
        accA = __builtin_amdgcn_wmma_f32_16x16x4_f32(
            false, aA, false, bA, (short)0, accA, false, false);
        accO = __builtin_amdgcn_wmma_f32_16x16x4_f32(
            false, aO, false, bO, (short)0, accO, false, false);
    }

    // D fragment: VGPR r -> lanes 0-15: M=r, N=lane; lanes 16-31: M=8+r, N=lane-16
    // Only columns N=0,1 carry real logits (lanes 0,1,16,17).
    if (row < 2) {
        const int mbase = khalf * 8;
#pragma unroll
        for (int r = 0; r < 8; ++r) {
            ep[wave][0][mbase + r][row] = accA[r];
            ep[wave][1][mbase + r][row] = accO[r];
        }
    }
    __syncthreads();

    if (lane < 16) {
        const int    tok = tok0 + lane;
        const size_t ts  = (size_t)b * SEQ + tok;
        const float  bA0 = base[b * 4 + 0];
        const float  bA1 = base[b * 4 + 1];
        const float  bO0 = base[b * 4 + 2];
        const float  bO1 = base[b * 4 + 3];

        float la0 = ep[wave][0][lane][0] + bA0;
        float la1 = ep[wave][0][lane][1] + bA1;
        float lo0 = ep[wave][1][lane][0] + bO0;
        float lo1 = ep[wave][1][lane][1] + bO1;

        const bool keep = (ign[ts] == 0);
        const bool va   = (mask_a[ts] == 1) && keep;
        const bool vo   = (mask_o[ts] == 1) && keep;

        float f0 = 0.5f * ((va ? la0 : 0.0f) + (vo ? lo0 : 0.0f));
        float f1 = 0.5f * ((va ? la1 : 0.0f) + (vo ? lo1 : 0.0f));

        v2f o;
        o.x = f0;
        o.y = f1;
        *(v2f*)(out_logits + ts * 2) = o;   // coalesced b64 stores

        const bool has  = (f0 + f1) != 0.0f;
        const int  lbl  = has ? (f1 > f0 ? 1 : 0) : -1;
        out_labels[ts]  = (float)lbl;
    }
}

// ---------------------------------------------------------------------------
// Launch
// ---------------------------------------------------------------------------
extern "C" void kernel_launch(void* const* d_in, const int* in_sizes, int n_in,
                              void* d_out, int out_size, void* d_ws, size_t ws_size,
                              hipStream_t stream) {
    const float* a2o_cls = (const float*)d_in[0];
    const float* a2o_hid = (const float*)d_in[1];
    const float* o2a_cls = (const float*)d_in[2];
    const float* o2a_hid = (const float*)d_in[3];
    const int*   mask_a  = (const int*)d_in[4];
    const int*   mask_o  = (const int*)d_in[5];
    const int*   ign     = (const int*)d_in[6];
    const float* W_a2o   = (const float*)d_in[7];
    const float* b_a2o   = (const float*)d_in[8];
    const float* W_o2a   = (const float*)d_in[9];
    const float* b_o2a   = (const float*)d_in[10];

    float* base       = (float*)d_ws;                       // [B][4] = 1 KB
    float* out_logits = (float*)d_out;                      // [B,S,2]
    float* out_labels = out_logits + (size_t)BATCH * SEQ * 2; // [B,S]

    cls_base_kernel<<<1, 256, 0, stream>>>(a2o_cls, o2a_cls,
                                           W_a2o, b_a2o, W_o2a, b_o2a, base);

    matching_wmma_kernel<<<GRID_BLOCKS, BLOCK_THREADS, 0, stream>>>(
        a2o_hid, o2a_hid, mask_a, mask_o, ign,
        W_a2o, W_o2a, base, out_logits, out_labels);
}